// CondenseMSA_4329327034442
// MI455X (gfx1250) — compile-verified
//
#include <hip/hip_runtime.h>

typedef __attribute__((ext_vector_type(2))) float v2f;
typedef __attribute__((ext_vector_type(8))) float v8f;

#define H 64
#define NF 16
#define EMB 48          // H - NF
#define NAA 21
#define BN_EPS 1e-5f

// ---------- workspace layout (4-byte words) ----------
// [0..31]     hist (int, 21 used)
// [32..47]    fsum   (per feature channel)
// [48..63]    fsumsq (per feature channel)
// [64..127]   b'  (fused bias)
// [128..4223] W't (64x64, [k][n] = W[n][k]*s[k])
// [4224..)    count (int, B*msl)
// [condOff..) condensed / avg_emb (R x 64 f32, GEMM done in place)

__global__ void k0_zero(float* out, int out_n, float* ws, int cntBase, int cntN) {
    int i = blockIdx.x * blockDim.x + threadIdx.x;
    if (i < out_n) out[i] = 0.0f;
    if (i < 64)    ws[i]  = 0.0f;
    if (i < cntN)  ((int*)ws)[cntBase + i] = 0;
}

// One pass over X and features:
//  - AA histogram (exact BN stats for embedding channels)
//  - sum/sumsq per feature channel (BN stats for feature channels)
//  - per-(b,t) mean over A of concat(E_aa[X], features) -> cond[R][64]
__global__ void k1_reduce(const int* __restrict__ X, const float* __restrict__ feat,
                          const float* __restrict__ Eaa, float* ws,
                          float* __restrict__ cond, int groups) {
    __shared__ float eaa[NAA * EMB];
    __shared__ int   xs[4 * 16];
    __shared__ int   histl[NAA];
    int tid = threadIdx.x;
    for (int i = tid; i < NAA * EMB; i += 256) eaa[i] = Eaa[i];
    if (tid < NAA) histl[tid] = 0;
    __syncthreads();

    int li = tid >> 6;        // 0..3  item-in-group
    int ch = tid & 63;        // 0..63 channel
    float accS = 0.0f, accS2 = 0.0f;

    for (int g = blockIdx.x; g < groups; g += gridDim.x) {
        int item = g * 4 + li;
        if (ch < 16) {                       // 16 threads stage X, bump hist
            int xv = X[item * 16 + ch];
            xs[li * 16 + ch] = xv;
            atomicAdd(&histl[xv], 1);
        }
        __syncthreads();
        float avg;
        if (ch < EMB) {                      // embedding channel: gather+avg
            float s = 0.0f;
            #pragma unroll
            for (int a = 0; a < 16; ++a) s += eaa[xs[li * 16 + a] * EMB + ch];
            avg = s * (1.0f / 16.0f);
        } else {                             // feature channel: avg + stats
            int c = ch - EMB;
            float s = 0.0f;
            const float* fp = feat + (size_t)item * 16 * NF + c;
            #pragma unroll
            for (int a = 0; a < 16; ++a) { float v = fp[a * NF]; s += v; accS2 += v * v; }
            accS += s;
            avg = s * (1.0f / 16.0f);
        }
        cond[(size_t)item * H + ch] = avg;
        __syncthreads();                     // protect xs for next group
    }
    if (ch >= EMB) {                         // one flush per thread
        atomicAdd(&ws[32 + (ch - EMB)], accS);
        atomicAdd(&ws[48 + (ch - EMB)], accS2);
    }
    if (tid < NAA) atomicAdd(&((int*)ws)[tid], histl[tid]);
}

// Fold BN into the linear layer: W't[k][n] = W[n][k]*s[k],  b'[n] = b[n] + sum_k t[k]*W[n][k]
__global__ void k2_fuse(const float* __restrict__ Eaa, const float* __restrict__ gamma,
                        const float* __restrict__ beta, const float* __restrict__ W,
                        const float* __restrict__ bias, float* ws, float Ninv) {
    __shared__ float sl[H], tl[H];
    int c = threadIdx.x;                     // 64 threads
    const int* hist = (const int*)ws;
    float mean, ex2;
    if (c < EMB) {                           // exact stats from histogram
        float m = 0.0f, m2 = 0.0f;
        for (int aa = 0; aa < NAA; ++aa) {
            float w = (float)hist[aa];
            float e = Eaa[aa * EMB + c];
            m += w * e; m2 += w * e * e;
        }
        mean = m * Ninv; ex2 = m2 * Ninv;
    } else {
        mean = ws[32 + c - EMB] * Ninv;
        ex2  = ws[48 + c - EMB] * Ninv;
    }
    float var = ex2 - mean * mean;
    float s = gamma[c] * rsqrtf(var + BN_EPS);
    sl[c] = s;
    tl[c] = beta[c] - mean * s;
    __syncthreads();
    int n = c;
    float bp = bias[n];
    for (int k = 0; k < H; ++k) {
        float w = W[n * H + k];
        bp += tl[k] * w;
        ws[128 + k * H + n] = w * sl[k];     // transposed, pre-scaled
    }
    ws[64 + n] = bp;
}

// cond[R][64] = cond[R][64] @ W't + b'   (in place; one wave per 16x64 tile)
// V_WMMA_F32_16X16X4_F32, K stepped by 4 (16 WMMAs per 16x16 accum).
__global__ void __launch_bounds__(256) k3_gemm(float* cond, const float* __restrict__ ws) {
    __shared__ float Wt[H * H];
    __shared__ float bp[H];
    int tid = threadIdx.x;
    for (int i = tid; i < H * H; i += 256) Wt[i] = ws[128 + i];
    if (tid < H) bp[tid] = ws[64 + tid];
    __syncthreads();

    int wave = tid >> 5, lane = tid & 31;
    int tile = blockIdx.x * 8 + wave;
    int lrow = lane & 15;
    int off  = (lane & 16) ? 2 : 0;          // K sub-offset per half-wave (ISA 7.12.2)

    const float* Arow = cond + (size_t)(tile * 16 + lrow) * H;
    v2f afr[16];                             // whole 16x64 A tile, reused over N-tiles
    #pragma unroll
    for (int kt = 0; kt < 16; ++kt)
        afr[kt] = *(const v2f*)(Arow + kt * 4 + off);

    #pragma unroll
    for (int nt = 0; nt < 4; ++nt) {
        int col = nt * 16 + lrow;
        float bb = bp[col];
        v8f acc = {bb, bb, bb, bb, bb, bb, bb, bb};   // bias pre-seeded into C
        #pragma unroll
        for (int kt = 0; kt < 16; ++kt) {
            v2f bfr;
            bfr.x = Wt[(kt * 4 + off) * H + col];
            bfr.y = Wt[(kt * 4 + off + 1) * H + col];
            acc = __builtin_amdgcn_wmma_f32_16x16x4_f32(
                false, afr[kt], false, bfr, (short)0, acc, false, false);
        }
        int rbase = tile * 16 + ((lane & 16) ? 8 : 0);
        #pragma unroll
        for (int r = 0; r < 8; ++r)
            cond[(size_t)(rbase + r) * H + col] = acc[r];
    }
}

__global__ void k4_scatter(const float* __restrict__ cond, const int* __restrict__ focuses,
                           float* out, int* cnt, int T, int msl) {
    int tid = threadIdx.x;
    int li = tid >> 6, ch = tid & 63;
    int item = blockIdx.x * 4 + li;
    int b = item / T;
    int f = focuses[item];
    atomicAdd(&out[((size_t)b * msl + f) * H + ch], cond[(size_t)item * H + ch]);
    if (ch == 0) atomicAdd(&cnt[b * msl + f], 1);
}

__global__ void k5_final(float* out, const int* __restrict__ cnt,
                         const int* __restrict__ seq_lens, int msl, int n) {
    int i = blockIdx.x * blockDim.x + threadIdx.x;
    if (i >= n) return;
    int pos = (i >> 6) % msl;
    int b   = i / (msl * H);
    int c = (pos >= seq_lens[b]) ? 1 : cnt[b * msl + pos];
    out[i] = out[i] / (float)c;              // 0/0 -> NaN, matching reference
}

extern "C" void kernel_launch(void* const* d_in, const int* in_sizes, int n_in,
                              void* d_out, int out_size, void* d_ws, size_t ws_size,
                              hipStream_t stream) {
    const int*   X        = (const int*)d_in[0];
    const float* feat     = (const float*)d_in[1];
    const int*   seq_lens = (const int*)d_in[2];
    const int*   focuses  = (const int*)d_in[3];
    const float* Eaa      = (const float*)d_in[8];
    const float* gamma    = (const float*)d_in[9];
    const float* beta     = (const float*)d_in[10];
    const float* W        = (const float*)d_in[11];
    const float* bias     = (const float*)d_in[12];
    float* out = (float*)d_out;
    float* ws  = (float*)d_ws;

    int B   = in_sizes[2];
    int R   = in_sizes[3];            // B*T rows after averaging over A
    int T   = R / B;
    int A   = in_sizes[0] / R;        // = 16
    int msl = out_size / (B * H);     // = 500

    int cntBase = 4224;
    int cntN    = B * msl;
    int condOff = ((cntBase + cntN + 255) / 256) * 256;
    float* cond = ws + condOff;
    int*   cnt  = (int*)ws + cntBase;

    int n0 = out_size > cntN ? out_size : cntN;
    k0_zero<<<(n0 + 255) / 256, 256, 0, stream>>>(out, out_size, ws, cntBase, cntN);

    int groups = R / 4;
    k1_reduce<<<240, 256, 0, stream>>>(X, feat, Eaa, ws, cond, groups);

    float Ninv = 1.0f / ((float)R * (float)A);
    k2_fuse<<<1, 64, 0, stream>>>(Eaa, gamma, beta, W, bias, ws, Ninv);

    k3_gemm<<<R / 128, 256, 0, stream>>>(cond, ws);     // 8 waves/block, 16 rows/wave

    k4_scatter<<<R / 4, 256, 0, stream>>>(cond, focuses, out, cnt, T, msl);

    k5_final<<<(out_size + 255) / 256, 256, 0, stream>>>(out, cnt, seq_lens, msl, out_size);
}